// PiCO_36679020708535
// MI455X (gfx1250) — compile-verified
//
#include <hip/hip_runtime.h>
#include <math.h>

// ---------------- problem constants ----------------
#define B_   4096
#define D_   128
#define C_   1000
#define Q_   65536
#define PM_  0.99f
#define EPS_ 1e-12f

// ---------------- output layout (float offsets) ----------------
#define OFF_CLASSFY 0ul                           // B*C          = 4,096,000
#define OFF_CLUSTER 4096000ul                     // B*C          = 4,096,000
#define OFF_CONTF   8192000ul                     // (2B+Q)*D     = 9,437,184
#define OFF_CONTL   17629184ul                    // 2B+Q         =    73,728
#define OFF_PROTO   17702912ul                    // C*D          =   128,000
#define OFF_NQUEUE  17830912ul                    // Q*D          = 8,388,608
#define OFF_NQP     26219520ul                    // Q            =    65,536
#define OFF_PTR     26285056ul                    // 1

typedef float v2f __attribute__((ext_vector_type(2)));
typedef float v8f __attribute__((ext_vector_type(8)));

// ============================================================
// 1) classifier softmax + pseudo labels (block per row)
//    writes classfy_out, and pl into cont_labels[0:B] and [B:2B]
// ============================================================
__global__ void __launch_bounds__(256)
k_classfy_softmax_label(const float* __restrict__ logits,
                        const float* __restrict__ plabel,
                        float* __restrict__ dout) {
    __shared__ float sred[256];
    __shared__ int   sidx[256];
    const int r = blockIdx.x;
    const int t = threadIdx.x;
    const float* x  = logits + (size_t)r * C_;
    const float* pl = plabel + (size_t)r * C_;
    float* co = dout + OFF_CLASSFY + (size_t)r * C_;

    // row max
    float mx = -3.402823466e38f;
    for (int c = t; c < C_; c += 256) mx = fmaxf(mx, x[c]);
    sred[t] = mx; __syncthreads();
    for (int s = 128; s > 0; s >>= 1) { if (t < s) sred[t] = fmaxf(sred[t], sred[t + s]); __syncthreads(); }
    mx = sred[0]; __syncthreads();

    // sum of exp
    float sm = 0.f;
    for (int c = t; c < C_; c += 256) sm += expf(x[c] - mx);
    sred[t] = sm; __syncthreads();
    for (int s = 128; s > 0; s >>= 1) { if (t < s) sred[t] += sred[t + s]; __syncthreads(); }
    const float inv = 1.0f / sred[0]; __syncthreads();

    // write probs + argmax(prob * plabel), first-index tie-break
    float bv = -1.0f; int bi = C_;
    for (int c = t; c < C_; c += 256) {
        float p = expf(x[c] - mx) * inv;
        co[c] = p;
        float s = p * pl[c];
        if (s > bv || (s == bv && c < bi)) { bv = s; bi = c; }
    }
    sred[t] = bv; sidx[t] = bi; __syncthreads();
    for (int s = 128; s > 0; s >>= 1) {
        if (t < s) {
            float v2 = sred[t + s]; int i2 = sidx[t + s];
            if (v2 > sred[t] || (v2 == sred[t] && i2 < sidx[t])) { sred[t] = v2; sidx[t] = i2; }
        }
        __syncthreads();
    }
    if (t == 0) {
        float plf = (float)sidx[0];
        dout[OFF_CONTL + r]      = plf;   // cont_labels[0:B]
        dout[OFF_CONTL + B_ + r] = plf;   // cont_labels[B:2B]
    }
}

// ============================================================
// 2) logits_prot = q @ prototypes^T via V_WMMA_F32_16X16X4_F32
//    one 16x16 tile per wave32; 63 N-tiles (tail masked, EXEC full)
// ============================================================
__global__ void __launch_bounds__(256)
k_gemm_wmma(const float* __restrict__ q, const float* __restrict__ prot,
            float* __restrict__ out /* = dout + OFF_CLUSTER, raw logits */) {
    const int NT = 63;                               // ceil(1000/16)
    const int wave = threadIdx.x >> 5;
    const int lane = threadIdx.x & 31;
    const int tile = blockIdx.x * 8 + wave;          // 2016 blocks * 8 waves = 16128 tiles
    const int mTile = tile / NT;
    const int nTile = tile % NT;
    const int mBase = mTile * 16;
    const int nBase = nTile * 16;

    const int lr = lane & 15;                        // row-within-tile lane index
    const int kh = lane >> 4;                        // K half-select (0: K=0,1 ; 1: K=2,3)

    // A: 16x4 fp32 tile of q. lane lr holds M=mBase+lr, K = kh*2 + {0,1}
    const int arow = mBase + lr;
    // B: 4x16 tile of prototypes^T => B[k][n] = prot[n][k]; lane lr holds N=nBase+lr
    const int brow  = nBase + lr;
    const float bscale = (brow < C_) ? 1.0f : 0.0f;  // mask tail, keep EXEC all-1s
    const int browc = (brow < C_) ? brow : (C_ - 1); // clamp => in-bounds load

    const float* aptr = q    + (size_t)arow  * D_ + kh * 2;
    const float* bptr = prot + (size_t)browc * D_ + kh * 2;

    v8f acc = {};
#pragma unroll 8
    for (int kb = 0; kb < D_; kb += 4) {
        v2f a, b;
        a.x = aptr[kb + 0];
        a.y = aptr[kb + 1];
        b.x = bptr[kb + 0] * bscale;
        b.y = bptr[kb + 1] * bscale;
        // D = A(16x4) * B(4x16) + C   (8 args: negA,A,negB,B,cmod,C,reuseA,reuseB)
        acc = __builtin_amdgcn_wmma_f32_16x16x4_f32(false, a, false, b,
                                                    (short)0, acc, false, false);
    }

    // C/D layout: VGPR j holds D[M = mBase + kh*8 + j][N = nBase + (lane&15)]
    const int col = nBase + lr;
    if (col < C_) {
        const int rbase = mBase + kh * 8;
#pragma unroll
        for (int j = 0; j < 8; ++j)
            out[(size_t)(rbase + j) * C_ + col] = acc[j];
    }
}

// ============================================================
// 3) in-place softmax over cluster logits region
// ============================================================
__global__ void __launch_bounds__(256)
k_cluster_softmax(float* __restrict__ dout) {
    __shared__ float sred[256];
    const int r = blockIdx.x;
    const int t = threadIdx.x;
    float* x = dout + OFF_CLUSTER + (size_t)r * C_;

    float mx = -3.402823466e38f;
    for (int c = t; c < C_; c += 256) mx = fmaxf(mx, x[c]);
    sred[t] = mx; __syncthreads();
    for (int s = 128; s > 0; s >>= 1) { if (t < s) sred[t] = fmaxf(sred[t], sred[t + s]); __syncthreads(); }
    mx = sred[0]; __syncthreads();

    float sm = 0.f;
    for (int c = t; c < C_; c += 256) sm += expf(x[c] - mx);
    sred[t] = sm; __syncthreads();
    for (int s = 128; s > 0; s >>= 1) { if (t < s) sred[t] += sred[t + s]; __syncthreads(); }
    const float inv = 1.0f / sred[0]; __syncthreads();

    for (int c = t; c < C_; c += 256) x[c] = expf(x[c] - mx) * inv;
}

// ============================================================
// 4) order-dependent EMA prototype update + L2 norm
//    block per class (c), thread per dim (d); labels cached in LDS
// ============================================================
__global__ void __launch_bounds__(128)
k_proto_ema(const float* __restrict__ q, const float* __restrict__ prot,
            float* __restrict__ dout) {
    __shared__ float labs[B_];    // 16 KB of the 320 KB WGP LDS
    __shared__ float red[128];
    const int c = blockIdx.x;
    const int t = threadIdx.x;    // dim index d

    const float* contl = dout + OFF_CONTL;           // pseudo labels from kernel 1
    for (int i = t; i < B_; i += 128) labs[i] = contl[i];
    __syncthreads();

    const float fc = (float)c;
    float v = prot[(size_t)c * D_ + t];
    for (int i = 0; i < B_; ++i) {                   // exact sequential semantics
        if (labs[i] == fc) v = PM_ * v + (1.0f - PM_) * q[(size_t)i * D_ + t];
    }

    red[t] = v * v; __syncthreads();
    for (int s = 64; s > 0; s >>= 1) { if (t < s) red[t] += red[t + s]; __syncthreads(); }
    const float n = fmaxf(sqrtf(red[0]), EPS_);
    dout[OFF_PROTO + (size_t)c * D_ + t] = v / n;
}

// ============================================================
// 5) cont_features = concat(q, k, queue)   (float4 streams)
// ============================================================
__global__ void __launch_bounds__(256)
k_contf(const float* __restrict__ q, const float* __restrict__ k,
        const float* __restrict__ queue, float* __restrict__ dout) {
    const size_t N4  = (size_t)(2 * B_ + Q_) * D_ / 4;
    const size_t BD4 = (size_t)B_ * D_ / 4;
    size_t i4 = (size_t)blockIdx.x * 256 + threadIdx.x;
    if (i4 >= N4) return;
    float4 val;
    if (i4 < BD4)            val = ((const float4*)q)[i4];
    else if (i4 < 2 * BD4)   val = ((const float4*)k)[i4 - BD4];
    else                     val = ((const float4*)queue)[i4 - 2 * BD4];
    ((float4*)(dout + OFF_CONTF))[i4] = val;
}

// ============================================================
// 6) new_queue = queue with rows [ptr, ptr+B) <- k (clamped splice)
// ============================================================
__global__ void __launch_bounds__(256)
k_newqueue(const float* __restrict__ k, const float* __restrict__ queue,
           const int* __restrict__ ptrp, float* __restrict__ dout) {
    const size_t N4 = (size_t)Q_ * D_ / 4;
    size_t i4 = (size_t)blockIdx.x * 256 + threadIdx.x;
    if (i4 >= N4) return;
    int ptr = *ptrp;                                  // dynamic_update_slice clamping
    ptr = ptr < 0 ? 0 : (ptr > Q_ - B_ ? Q_ - B_ : ptr);
    const size_t row = i4 / (D_ / 4);
    const size_t cch = i4 % (D_ / 4);
    float4 val;
    if (row >= (size_t)ptr && row < (size_t)ptr + B_)
        val = ((const float4*)k)[(row - (size_t)ptr) * (D_ / 4) + cch];
    else
        val = ((const float4*)queue)[i4];
    ((float4*)(dout + OFF_NQUEUE))[i4] = val;
}

// ============================================================
// 7) cont_labels tail, new_queue_pseudo splice, new_ptr
// ============================================================
__global__ void __launch_bounds__(256)
k_tail(const float* __restrict__ queue_pseudo, const int* __restrict__ ptrp,
       float* __restrict__ dout) {
    const int j = blockIdx.x * 256 + threadIdx.x;
    if (j >= Q_) return;
    const float qp = queue_pseudo[j];
    dout[OFF_CONTL + 2 * B_ + j] = qp;                // cont_labels[2B:]
    int ptr = *ptrp;
    ptr = ptr < 0 ? 0 : (ptr > Q_ - B_ ? Q_ - B_ : ptr);
    float v = qp;
    if (j >= ptr && j < ptr + B_)
        v = dout[OFF_CONTL + (size_t)(j - ptr)];      // pl written by kernel 1
    dout[OFF_NQP + j] = v;
    if (j == 0) dout[OFF_PTR] = (float)(((*ptrp) + B_) % Q_);
}

// ============================================================
extern "C" void kernel_launch(void* const* d_in, const int* in_sizes, int n_in,
                              void* d_out, int out_size, void* d_ws, size_t ws_size,
                              hipStream_t stream) {
    (void)in_sizes; (void)n_in; (void)out_size; (void)d_ws; (void)ws_size;
    const float* q      = (const float*)d_in[0];
    const float* k      = (const float*)d_in[1];
    const float* logits = (const float*)d_in[2];
    const float* plabel = (const float*)d_in[3];
    const float* prot   = (const float*)d_in[4];
    const float* queue  = (const float*)d_in[5];
    const float* qps    = (const float*)d_in[6];
    const int*   ptrp   = (const int*)d_in[7];
    float* dout = (float*)d_out;

    // 1) classifier softmax + pseudo labels (labels land in cont_labels slots)
    k_classfy_softmax_label<<<B_, 256, 0, stream>>>(logits, plabel, dout);
    // 2) WMMA GEMM: raw cluster logits into cluster_out region
    k_gemm_wmma<<<2016, 256, 0, stream>>>(q, prot, dout + OFF_CLUSTER);
    // 3) softmax in place over cluster logits
    k_cluster_softmax<<<B_, 256, 0, stream>>>(dout);
    // 4) sequential-order EMA prototype update + L2 norm
    k_proto_ema<<<C_, 128, 0, stream>>>(q, prot, dout);
    // 5) cont_features concat
    {
        const size_t n4 = (size_t)(2 * B_ + Q_) * D_ / 4;
        k_contf<<<(unsigned)((n4 + 255) / 256), 256, 0, stream>>>(q, k, queue, dout);
    }
    // 6) new_queue splice
    {
        const size_t n4 = (size_t)Q_ * D_ / 4;
        k_newqueue<<<(unsigned)((n4 + 255) / 256), 256, 0, stream>>>(k, queue, ptrp, dout);
    }
    // 7) label tails + new_ptr
    k_tail<<<(Q_ + 255) / 256, 256, 0, stream>>>(qps, ptrp, dout);
}